// LFModel_86638080295365
// MI455X (gfx1250) — compile-verified
//
#include <hip/hip_runtime.h>
#include <cstddef>
#include <cstdint>

#define D  192
#define S  4096
#define NE 8

typedef __attribute__((ext_vector_type(2))) float v2f;
typedef __attribute__((ext_vector_type(4))) float v4f;
typedef __attribute__((ext_vector_type(8))) float v8f;

// -----------------------------------------------------------------------------
// Fused GEMM: out[s, j] = sum_k A[s,k] * W[j,k] + bias[j], optional per-row
// LayerNorm (gamma/beta) and ReLU. Block = 128 threads (4 waves) handles 64
// rows x all 192 cols. Weight (147KB) + A tile (padded 64x196, 48KB) staged in
// dynamic LDS (~193KB <= 320KB/WGP). Core math: v_wmma_f32_16x16x4_f32.
// -----------------------------------------------------------------------------
__global__ __launch_bounds__(128)
void gemm_rows_fused(const float* __restrict__ A, const float* __restrict__ W,
                     const float* __restrict__ bias, float* __restrict__ out,
                     const float* __restrict__ ln_g, const float* __restrict__ ln_b,
                     int do_relu) {
  extern __shared__ float smem[];
  float* sW = smem;            // D*D = 36864 floats
  float* sA = smem + D * D;    // 64 rows, stride 196 (pad vs bank conflicts)

  const int tid  = threadIdx.x;
  const int row0 = blockIdx.x * 64;

  // Stage weight matrix (row-major [j][k]) into LDS, coalesced float4.
  {
    const v4f* Wv  = (const v4f*)W;
    v4f*       sWv = (v4f*)sW;
    for (int i = tid; i < (D * D) / 4; i += 128) sWv[i] = Wv[i];
  }
  // Stage A tile: 64 x 192 -> padded stride 196.
  {
    const v4f* Av = (const v4f*)(A + (size_t)row0 * D);
    for (int i = tid; i < 64 * 48; i += 128) {
      int r = i / 48, c = i % 48;
      *(v4f*)&sA[r * 196 + c * 4] = Av[r * 48 + c];
    }
  }
  __syncthreads();

  const int wave = tid >> 5;
  const int lane = tid & 31;
  const int half = lane >> 4;   // lane group: 0 => lanes 0-15, 1 => 16-31
  const int l16  = lane & 15;

  // Per-lane bias for the 12 N-tiles (col = nt*16 + l16).
  float breg[12];
#pragma unroll
  for (int nt = 0; nt < 12; ++nt) breg[nt] = bias[nt * 16 + l16];

  v8f acc[12];
#pragma unroll
  for (int nt = 0; nt < 12; ++nt)
#pragma unroll
    for (int e = 0; e < 8; ++e) acc[nt][e] = 0.0f;

  // A fragment row for this lane (A layout: lane16 = M row; VGPR pair holds
  // K = kb + 2*half + {0,1}).  B fragment: lane16 = N col, same K pairing;
  // since B = W^T, b-pair loads W[col][kb + 2*half + {0,1}].
  const float* aRow = &sA[(wave * 16 + l16) * 196];
  for (int kb = 0; kb < D; kb += 4) {
    v2f af = *(const v2f*)&aRow[kb + 2 * half];
#pragma unroll
    for (int nt = 0; nt < 12; ++nt) {
      v2f bf = *(const v2f*)&sW[(size_t)(nt * 16 + l16) * D + kb + 2 * half];
      acc[nt] = __builtin_amdgcn_wmma_f32_16x16x4_f32(
          false, af, false, bf, (short)0, acc[nt], false, false);
    }
  }
  __syncthreads();  // all waves done with k-loop before sA is reused

  // Stage C (+bias) into sA: D layout -> acc[nt][r] = C[M = r + 8*half][N=l16].
#pragma unroll
  for (int nt = 0; nt < 12; ++nt) {
#pragma unroll
    for (int r = 0; r < 8; ++r) {
      sA[(size_t)(wave * 16 + r + 8 * half) * 196 + nt * 16 + l16] =
          acc[nt][r] + breg[nt];
    }
  }
  __syncthreads();

  // Epilogue: one thread per row -> optional LayerNorm + ReLU, store.
  if (tid < 64) {
    float* rowp = &sA[(size_t)tid * 196];
    float* orow = out + (size_t)(row0 + tid) * D;
    if (ln_g != nullptr) {
      float m = 0.0f;
      for (int k = 0; k < D; ++k) m += rowp[k];
      m *= (1.0f / D);
      float v = 0.0f;
      for (int k = 0; k < D; ++k) { float d = rowp[k] - m; v += d * d; }
      v *= (1.0f / D);
      float rs = rsqrtf(v + 1e-5f);
      for (int k = 0; k < D; ++k) {
        float val = (rowp[k] - m) * rs * ln_g[k] + ln_b[k];
        if (do_relu) val = fmaxf(val, 0.0f);
        orow[k] = val;
      }
    } else {
      for (int k = 0; k < D; ++k) {
        float val = rowp[k];
        if (do_relu) val = fmaxf(val, 0.0f);
        orow[k] = val;
      }
    }
  }
}

// -----------------------------------------------------------------------------
// Column mean over S: xm[c] = mean_s h[s, c].
// -----------------------------------------------------------------------------
__global__ __launch_bounds__(256)
void colmean_kernel(const float* __restrict__ h, float* __restrict__ xm) {
  __shared__ float red[256];
  const int c = blockIdx.x, t = threadIdx.x;
  float s = 0.0f;
  for (int r = t; r < S; r += 256) s += h[(size_t)r * D + c];
  red[t] = s;
  __syncthreads();
  for (int off = 128; off > 0; off >>= 1) {
    if (t < off) red[t] += red[t + off];
    __syncthreads();
  }
  if (t == 0) xm[c] = red[0] * (1.0f / (float)S);
}

// -----------------------------------------------------------------------------
// Adapted weight stream: Wout[j,k] = Wbase[j,k] + Ab[j*D+k] + AW[j*D+k, :]@vec.
// Grid = D*3 blocks; block handles j = bid/3 and 64 consecutive k. The 64x192
// AW slab is contiguous in memory -> fully coalesced float4 stream into LDS;
// 4 threads cooperate per output element. This kernel is the HBM-bound part
// of the model (AW matrices total ~85MB across feat/tm/cm).
// -----------------------------------------------------------------------------
__global__ __launch_bounds__(256)
void adaptw_kernel(const float* __restrict__ AW, const float* __restrict__ Ab,
                   const float* __restrict__ Wbase, const float* __restrict__ vec,
                   float* __restrict__ Wout) {
  __shared__ float sa[D];
  __shared__ float tile[64 * D];  // 48KB
  __shared__ float red[256];
  const int t  = threadIdx.x;
  const int j  = blockIdx.x / 3;
  const int kt = (blockIdx.x % 3) * 64;

  if (t < D) sa[t] = vec[t];
  {
    const v4f* src = (const v4f*)(AW + ((size_t)j * D + kt) * D);
    v4f*       dst = (v4f*)tile;
    for (int i = t; i < (64 * D) / 4; i += 256) dst[i] = src[i];
  }
  __syncthreads();

  const int o = t >> 2, p = t & 3;
  float s = 0.0f;
  for (int l = p * 48; l < p * 48 + 48; ++l) s += tile[o * D + l] * sa[l];
  red[t] = s;
  __syncthreads();
  if (p == 0) {
    size_t idx = (size_t)j * D + kt + o;
    float d = red[t] + red[t + 1] + red[t + 2] + red[t + 3];
    Wout[idx] = Wbase[idx] + Ab[idx] + d;
  }
}

// -----------------------------------------------------------------------------
// Gate-combined expert weight: Wce[j,k] = sum_n gate[n] * (ex_W[n,j,k]
//   + ex_Ab[n,j*D+k] + ex_AW[n,j*D+k,:]@a0).  Streams the full 226MB ex_AW
// exactly once -> this kernel sets the kernel-chain's roofline (~10us @ 23.3TB/s).
// -----------------------------------------------------------------------------
__global__ __launch_bounds__(256)
void adaptw_experts_kernel(const float* __restrict__ AW, const float* __restrict__ Ab,
                           const float* __restrict__ Wbase, const float* __restrict__ vec,
                           const float* __restrict__ gate, float* __restrict__ Wout) {
  __shared__ float sa[D];
  __shared__ float sg[NE];
  __shared__ float tile[64 * D];
  __shared__ float red[256];
  const int t  = threadIdx.x;
  const int j  = blockIdx.x / 3;
  const int kt = (blockIdx.x % 3) * 64;

  if (t < D)  sa[t] = vec[t];
  if (t < NE) sg[t] = gate[t];

  const int o = t >> 2, p = t & 3;
  float acc = 0.0f;
  for (int n = 0; n < NE; ++n) {
    __syncthreads();  // tile reuse fence (also publishes sa/sg on n==0)
    const v4f* src =
        (const v4f*)(AW + (size_t)n * D * D * D + ((size_t)j * D + kt) * D);
    v4f* dst = (v4f*)tile;
    for (int i = t; i < (64 * D) / 4; i += 256) dst[i] = src[i];
    __syncthreads();
    float s = 0.0f;
    for (int l = p * 48; l < p * 48 + 48; ++l) s += tile[o * D + l] * sa[l];
    acc += sg[n] * s;
  }
  red[t] = acc;
  __syncthreads();
  if (p == 0) {
    size_t idx = (size_t)j * D + kt + o;
    float d = red[t] + red[t + 1] + red[t + 2] + red[t + 3];
    float base = 0.0f;
    for (int n = 0; n < NE; ++n)
      base += sg[n] * (Wbase[(size_t)n * D * D + idx] + Ab[(size_t)n * D * D + idx]);
    Wout[idx] = base + d;
  }
}

// -----------------------------------------------------------------------------
// a_raw[j] = bias[j] + sum_k vec[k] * Wf[j,k]   (192x192 matvec, 192 blocks)
// -----------------------------------------------------------------------------
__global__ __launch_bounds__(D)
void matvec192_kernel(const float* __restrict__ Wf, const float* __restrict__ vec,
                      const float* __restrict__ bias, float* __restrict__ out) {
  __shared__ float sv[D];
  __shared__ float red[D];
  const int j = blockIdx.x, t = threadIdx.x;
  sv[t] = vec[t];
  __syncthreads();
  red[t] = sv[t] * Wf[(size_t)j * D + t];
  __syncthreads();
  if (t == 0) {
    float s = 0.0f;
    for (int k = 0; k < D; ++k) s += red[k];
    out[j] = s + bias[j];
  }
}

// -----------------------------------------------------------------------------
// Single tiny block: LN(a_raw) -> a0; gate = softmax(a0 @ gate_W^T + gate_b);
// bce[j] = sum_n gate[n]*ex_b[n,j].
// -----------------------------------------------------------------------------
__global__ __launch_bounds__(256)
void finalize_kernel(const float* __restrict__ a_raw, const float* __restrict__ g,
                     const float* __restrict__ b, const float* __restrict__ gate_W,
                     const float* __restrict__ gate_b, const float* __restrict__ ex_b,
                     float* __restrict__ a0, float* __restrict__ gate,
                     float* __restrict__ bce) {
  __shared__ float sa[D];
  __shared__ float sn[D];
  __shared__ float sg[NE];
  __shared__ float stats[2];
  const int t = threadIdx.x;
  if (t < D) sa[t] = a_raw[t];
  __syncthreads();
  if (t == 0) {
    float m = 0.0f;
    for (int k = 0; k < D; ++k) m += sa[k];
    m *= (1.0f / D);
    float v = 0.0f;
    for (int k = 0; k < D; ++k) { float d = sa[k] - m; v += d * d; }
    v *= (1.0f / D);
    stats[0] = m;
    stats[1] = rsqrtf(v + 1e-5f);
  }
  __syncthreads();
  if (t < D) {
    float val = (sa[t] - stats[0]) * stats[1] * g[t] + b[t];
    sn[t] = val;
    a0[t] = val;
  }
  __syncthreads();
  if (t < NE) {
    float s = gate_b[t];
    for (int k = 0; k < D; ++k) s += gate_W[(size_t)t * D + k] * sn[k];
    sg[t] = s;
  }
  __syncthreads();
  if (t == 0) {
    float mx = sg[0];
    for (int n = 1; n < NE; ++n) mx = fmaxf(mx, sg[n]);
    float ss = 0.0f;
    for (int n = 0; n < NE; ++n) { sg[n] = __expf(sg[n] - mx); ss += sg[n]; }
    float inv = 1.0f / ss;
    for (int n = 0; n < NE; ++n) { sg[n] *= inv; gate[n] = sg[n]; }
  }
  __syncthreads();
  if (t < D) {
    float s = 0.0f;
    for (int n = 0; n < NE; ++n) s += sg[n] * ex_b[(size_t)n * D + t];
    bce[t] = s;
  }
}

// -----------------------------------------------------------------------------
extern "C" void kernel_launch(void* const* d_in, const int* in_sizes, int n_in,
                              void* d_out, int out_size, void* d_ws, size_t ws_size,
                              hipStream_t stream) {
  (void)in_sizes; (void)n_in; (void)out_size; (void)ws_size;

  const float* x       = (const float*)d_in[0];
  const float* W_in    = (const float*)d_in[1];
  const float* b_in    = (const float*)d_in[2];
  const float* feat_W  = (const float*)d_in[3];
  const float* feat_b  = (const float*)d_in[4];
  const float* feat_AW = (const float*)d_in[5];
  const float* feat_Ab = (const float*)d_in[6];
  const float* an_g    = (const float*)d_in[7];
  const float* an_b    = (const float*)d_in[8];
  const float* tm_W    = (const float*)d_in[9];
  const float* tm_b    = (const float*)d_in[10];
  const float* tm_AW   = (const float*)d_in[11];
  const float* tm_Ab   = (const float*)d_in[12];
  const float* tn_g    = (const float*)d_in[13];
  const float* tn_b    = (const float*)d_in[14];
  const float* cm_W    = (const float*)d_in[15];
  const float* cm_b    = (const float*)d_in[16];
  const float* cm_AW   = (const float*)d_in[17];
  const float* cm_Ab   = (const float*)d_in[18];
  const float* cn_g    = (const float*)d_in[19];
  const float* cn_b    = (const float*)d_in[20];
  const float* pm_W    = (const float*)d_in[21];
  const float* pm_b    = (const float*)d_in[22];
  const float* gate_W  = (const float*)d_in[23];
  const float* gate_b  = (const float*)d_in[24];
  const float* ex_W    = (const float*)d_in[25];
  const float* ex_b    = (const float*)d_in[26];
  const float* ex_AW   = (const float*)d_in[27];
  const float* ex_Ab   = (const float*)d_in[28];
  const float* mn_g    = (const float*)d_in[29];
  const float* mn_b    = (const float*)d_in[30];
  const float* out_W   = (const float*)d_in[31];
  const float* out_b   = (const float*)d_in[32];

  float* ws    = (float*)d_ws;
  float* buf0  = ws;                     // h, later m      (S*D)
  float* buf1  = ws + (size_t)S * D;     // t, later eo     (S*D)
  float* buf2  = ws + 2 * (size_t)S * D; // c               (S*D)
  float* Wt    = ws + 3 * (size_t)S * D; // D*D
  float* Wc    = Wt + D * D;             // D*D
  float* Wce   = Wc + D * D;             // D*D (temp Wf, then combined experts)
  float* xm    = Wce + D * D;            // D
  float* a_raw = xm + D;                 // D
  float* a0v   = a_raw + D;              // D
  float* gate  = a0v + D;                // NE
  float* bce   = gate + NE;              // D

  const size_t shmem = (size_t)(D * D + 64 * 196) * sizeof(float);

  // 1) h = x @ W_in^T + b_in
  gemm_rows_fused<<<S / 64, 128, shmem, stream>>>(x, W_in, b_in, buf0,
                                                  nullptr, nullptr, 0);
  // 2) xm = mean_s h
  colmean_kernel<<<D, 256, 0, stream>>>(buf0, xm);
  // 3) Wf = feat_W + reshape(feat_AW @ xm + feat_Ab)   (Wce used as temp)
  adaptw_kernel<<<D * 3, 256, 0, stream>>>(feat_AW, feat_Ab, feat_W, xm, Wce);
  // 4) a_raw = Wf @ xm + feat_b
  matvec192_kernel<<<D, D, 0, stream>>>(Wce, xm, feat_b, a_raw);
  // 5) a0 = LN(a_raw); gate = softmax(a0 @ gate_W^T + gate_b); bce = gate.ex_b
  finalize_kernel<<<1, 256, 0, stream>>>(a_raw, an_g, an_b, gate_W, gate_b,
                                         ex_b, a0v, gate, bce);
  // 6/7) adapted token-mix / channel-mix weights
  adaptw_kernel<<<D * 3, 256, 0, stream>>>(tm_AW, tm_Ab, tm_W, a0v, Wt);
  adaptw_kernel<<<D * 3, 256, 0, stream>>>(cm_AW, cm_Ab, cm_W, a0v, Wc);
  // 8) gate-combined expert weight (streams full 226MB ex_AW once)
  adaptw_experts_kernel<<<D * 3, 256, 0, stream>>>(ex_AW, ex_Ab, ex_W, a0v,
                                                   gate, Wce);
  // 9)  t = LN(h @ Wt^T + tm_b)
  gemm_rows_fused<<<S / 64, 128, shmem, stream>>>(buf0, Wt, tm_b, buf1,
                                                  tn_g, tn_b, 0);
  // 10) c = LN(t @ Wc^T + cm_b)
  gemm_rows_fused<<<S / 64, 128, shmem, stream>>>(buf1, Wc, cm_b, buf2,
                                                  cn_g, cn_b, 0);
  // 11) m = c @ pm_W^T + pm_b
  gemm_rows_fused<<<S / 64, 128, shmem, stream>>>(buf2, pm_W, pm_b, buf0,
                                                  nullptr, nullptr, 0);
  // 12) eo = relu(LN(m @ Wce^T + bce))
  gemm_rows_fused<<<S / 64, 128, shmem, stream>>>(buf0, Wce, bce, buf1,
                                                  mn_g, mn_b, 1);
  // 13) out = eo @ out_W^T + out_b
  gemm_rows_fused<<<S / 64, 128, shmem, stream>>>(buf1, out_W, out_b,
                                                  (float*)d_out, nullptr,
                                                  nullptr, 0);
}